// KeyValueCaching_56667798503562
// MI455X (gfx1250) — compile-verified
//
#include <hip/hip_runtime.h>
#include <hip/hip_bf16.h>

// KV-cache append: out = concat(cache, proj) along seq axis, for K and V.
// Pure bandwidth problem: ~1.07 GB HBM traffic -> ~46 us floor at 23.3 TB/s.
// No FLOPs -> no WMMA; the CDNA5 paths that matter are NT-hinted b128
// streaming and the ASYNCcnt-tracked async global<->LDS data mover.

#define KV_B 8
#define KV_S 4096
#define KV_D 2048

typedef float __attribute__((ext_vector_type(4))) f4;
typedef int   __attribute__((ext_vector_type(4))) i4;

// ---------------------------------------------------------------------------
// Bulk copy: grid (1024, B, 2). Each block copies a contiguous 32 KB tile
// (2048 f4) of cache[b] into out rows [0,S). Fully static iteration space:
// 8 b128 NT loads + 8 b128 NT stores per thread, immediate offsets j*4096 B
// off one address pair -> no bounds checks, no per-iteration address math.
// ---------------------------------------------------------------------------
__global__ __launch_bounds__(256) void kv_bulk_copy(const float* __restrict__ k_cache,
                                                    const float* __restrict__ v_cache,
                                                    float* __restrict__ out) {
    const unsigned b = blockIdx.y;   // batch
    const unsigned z = blockIdx.z;   // 0 = K, 1 = V

    const float* src = (z == 0) ? k_cache : v_cache;
    const size_t per_out = (size_t)KV_B * (KV_S + 1) * KV_D;   // floats per output tensor

    // Block tile: 2048 f4 = 32 KB, contiguous. 1024 blocks cover S*D/4 exactly.
    const size_t tile = (size_t)blockIdx.x * 2048 + threadIdx.x;   // f4 units

    const f4* __restrict__ s4 = (const f4*)(src + (size_t)b * KV_S * KV_D) + tile;
    f4* __restrict__ d4 = (f4*)(out + (size_t)z * per_out
                                    + (size_t)b * (KV_S + 1) * KV_D) + tile;

#pragma unroll
    for (int j = 0; j < 8; ++j) {
        // offset j*256 f4 = j*4096 bytes: fits the 24-bit immediate, so this
        // lowers to global_load_b128/global_store_b128 ... offset:j*4096 th:NT
        f4 v = __builtin_nontemporal_load(s4 + j * 256);
        __builtin_nontemporal_store(v, d4 + j * 256);
    }
}

// ---------------------------------------------------------------------------
// Tail copy: row s == S comes from k_proj / v_proj (D floats per (b,z)).
// Uses the CDNA5 async data-mover path (global <-> LDS async b128 ops,
// ASYNCcnt + s_wait_asynccnt). Each lane stages only its own 16B chunks, so
// the per-wave ASYNCcnt wait is the only synchronization needed.
// ---------------------------------------------------------------------------
#if __has_builtin(__builtin_amdgcn_global_load_async_to_lds_b128) && \
    __has_builtin(__builtin_amdgcn_global_store_async_from_lds_b128) && \
    __has_builtin(__builtin_amdgcn_s_wait_asynccnt)
#define KV_HAVE_ASYNC_LDS 1
#else
#define KV_HAVE_ASYNC_LDS 0
#endif

__global__ __launch_bounds__(256) void kv_tail_copy(const float* __restrict__ k_proj,
                                                    const float* __restrict__ v_proj,
                                                    float* __restrict__ out) {
    const unsigned b = blockIdx.x;   // batch
    const unsigned z = blockIdx.y;   // 0 = K, 1 = V

    const float* src = ((z == 0) ? k_proj : v_proj) + (size_t)b * KV_D;
    const size_t per_out = (size_t)KV_B * (KV_S + 1) * KV_D;
    float* dst = out + (size_t)z * per_out + (size_t)b * (KV_S + 1) * KV_D + (size_t)KV_S * KV_D;

    const int t = threadIdx.x;  // 256 threads; D = 2048 floats = 512 f4 -> 2 f4/lane

#if KV_HAVE_ASYNC_LDS
    __shared__ __align__(16) float lds_stage[KV_D];

    // Builtin signature (probe-confirmed): (int4 AS1*, int4 AS3*, Ii off, Ii cpol)
    auto* gsrc = (__attribute__((address_space(1))) i4*)const_cast<float*>(src);
    auto* gdst = (__attribute__((address_space(1))) i4*)dst;
    auto* lptr = (__attribute__((address_space(3))) i4*)lds_stage;

#pragma unroll
    for (int j = 0; j < 2; ++j) {
        const int idx = t + j * 256;                           // 16B units
        __builtin_amdgcn_global_load_async_to_lds_b128(
            gsrc + idx, lptr + idx, /*imm offset*/ 0, /*cpol*/ 0);
    }
    __builtin_amdgcn_s_wait_asynccnt(0);   // loads into LDS complete (per-wave)

#pragma unroll
    for (int j = 0; j < 2; ++j) {
        const int idx = t + j * 256;
        __builtin_amdgcn_global_store_async_from_lds_b128(
            gdst + idx, lptr + idx, /*imm offset*/ 0, /*cpol*/ 0);
    }
    __builtin_amdgcn_s_wait_asynccnt(0);   // stores handed to memory
#else
    const f4* __restrict__ s4 = (const f4*)src;
    f4* __restrict__ d4 = (f4*)dst;
#pragma unroll
    for (int j = 0; j < 2; ++j) {
        const int i4x = t + j * 256;
        f4 v = __builtin_nontemporal_load(&s4[i4x]);
        __builtin_nontemporal_store(v, &d4[i4x]);
    }
#endif
}

// ---------------------------------------------------------------------------
// Launch: inputs in setup_inputs() order: k_cache, v_cache, k_proj, v_proj.
// d_out = [k_new | v_new] flat, 2 * B*(S+1)*D floats. Disjoint writes; every
// output element is written exactly once (rows [0,S) by bulk, row S by tail).
// ---------------------------------------------------------------------------
extern "C" void kernel_launch(void* const* d_in, const int* in_sizes, int n_in,
                              void* d_out, int out_size, void* d_ws, size_t ws_size,
                              hipStream_t stream) {
    (void)in_sizes; (void)n_in; (void)out_size; (void)d_ws; (void)ws_size;

    const float* k_cache = (const float*)d_in[0];
    const float* v_cache = (const float*)d_in[1];
    const float* k_proj  = (const float*)d_in[2];
    const float* v_proj  = (const float*)d_in[3];
    float* out = (float*)d_out;

    // Bulk: 1024 x 8 x 2 blocks, 256 threads (8 wave32s) each; 16384 blocks /
    // 32 KB tiles -> plenty of in-flight b128s to saturate 23.3 TB/s.
    dim3 bulk_grid(1024, KV_B, 2);
    kv_bulk_copy<<<bulk_grid, 256, 0, stream>>>(k_cache, v_cache, out);

    // Tail: one 256-thread block per (batch, tensor) writes row s == S.
    dim3 tail_grid(KV_B, 2);
    kv_tail_copy<<<tail_grid, 256, 0, stream>>>(k_proj, v_proj, out);
}